// LinModel_6373731467824
// MI455X (gfx1250) — compile-verified
//
#include <hip/hip_runtime.h>

// ---------------------------------------------------------------------------
// out[b,h] = x[b,:] . W[h,:] + bias[h]   (B x 21 GEMM, K=1024)
// result[b] = sum_h out[b,h] * v_conf[b,h]
// d_out = [ result (B floats) | out (B*21 floats) ]
//
// Memory-bound kernel (x stream = 512MB @ 23.3TB/s ~= 23us floor).
// Uses V_WMMA_F32_16X16X4_F32 (exact f32) with W staged in LDS.
// ---------------------------------------------------------------------------

typedef float v2f __attribute__((ext_vector_type(2)));
typedef float v8f __attribute__((ext_vector_type(8)));

#define DIM   1024
#define NH    21
#define WROW  1026          // LDS row stride in floats: 8B aligned, bank-stride 2
#define WAVES 8
#define ROWS_PER_WAVE 16
#define ROWS_PER_BLOCK (WAVES * ROWS_PER_WAVE)   // 128

__global__ __launch_bounds__(256, 1)
void linheads_wmma_kernel(const float* __restrict__ x,
                          const float* __restrict__ vconf,
                          const float* __restrict__ W,
                          const float* __restrict__ bias,
                          float* __restrict__ result,
                          float* __restrict__ out)
{
    // W staged in LDS: rows 0..20 real, row 21 zeroed (clamp target for h>=21)
    __shared__ float Wlds[22 * WROW];

    const int tid = threadIdx.x;

    for (int i = tid; i < NH * DIM; i += 256) {
        int h = i >> 10;        // / 1024
        int k = i & (DIM - 1);  // % 1024
        Wlds[h * WROW + k] = W[i];
    }
    for (int i = tid; i < WROW; i += 256) Wlds[21 * WROW + i] = 0.0f;
    __syncthreads();

    const int lane = tid & 31;
    const int wv   = tid >> 5;
    const int l16  = lane & 15;     // column within N-tile / row within M-tile
    const int hi   = lane >> 4;     // 0 = lanes 0-15, 1 = lanes 16-31
    const int koff = hi * 2;        // K phase for A/B fragments (16x16x4 layout)

    const int m0 = (blockIdx.x * WAVES + wv) * ROWS_PER_WAVE;

    // A fragment source: lane holds row m0+l16, K pair {k+koff, k+koff+1}
    const float* Ap = x + (size_t)(m0 + l16) * DIM + koff;

    // B fragment sources (LDS): lane holds column h = h0+l16, same K pair.
    // Tile 0: h = 0..15 (all valid).  Tile 1: h = 16..31, clamp >=21 to zero row.
    int h1 = 16 + l16;
    if (h1 > 21) h1 = 21;           // zero row
    const float* Bp0 = &Wlds[l16 * WROW + koff];
    const float* Bp1 = &Wlds[h1  * WROW + koff];

    v8f acc0 = {0.f, 0.f, 0.f, 0.f, 0.f, 0.f, 0.f, 0.f};
    v8f acc1 = {0.f, 0.f, 0.f, 0.f, 0.f, 0.f, 0.f, 0.f};

#pragma unroll 8
    for (int k = 0; k < DIM; k += 4) {
        v2f a  = *(const v2f*)(Ap  + k);
        v2f b0 = *(const v2f*)(Bp0 + k);
        v2f b1 = *(const v2f*)(Bp1 + k);
        acc0 = __builtin_amdgcn_wmma_f32_16x16x4_f32(
                   false, a, false, b0, (short)0, acc0, false, false);
        acc1 = __builtin_amdgcn_wmma_f32_16x16x4_f32(
                   false, a, false, b1, (short)0, acc1, false, false);
    }

    // ---- epilogue: bias, store out, fused v_conf reduction ----
    const float bia0 = bias[l16];                          // h = l16 < 16
    const float bia1 = (l16 < 5) ? bias[16 + l16] : 0.0f;  // h = 16..20

    float p[8];
#pragma unroll
    for (int j = 0; j < 8; ++j) {
        const int m = m0 + j + hi * 8;     // C/D layout: VGPR j -> row j (+8 upper)
        const size_t rowBase = (size_t)m * NH;

        float o0 = acc0[j] + bia0;
        float o1 = acc1[j] + bia1;

        out[rowBase + l16] = o0;
        float vc0 = vconf[rowBase + l16];
        float vc1 = 0.0f;
        if (l16 < 5) {
            out[rowBase + 16 + l16] = o1;
            vc1 = vconf[rowBase + 16 + l16];
        }
        p[j] = o0 * vc0 + o1 * vc1;
    }

    // Butterfly-reduce over the 16 column lanes of each half-wave (wave32).
#pragma unroll
    for (int j = 0; j < 8; ++j) {
        p[j] += __shfl_xor(p[j], 1, 16);
        p[j] += __shfl_xor(p[j], 2, 16);
        p[j] += __shfl_xor(p[j], 4, 16);
        p[j] += __shfl_xor(p[j], 8, 16);
    }

    // Lane l16==j (both halves) writes result for row m0 + hi*8 + j.
#pragma unroll
    for (int j = 0; j < 8; ++j) {
        if (l16 == j) result[m0 + hi * 8 + j] = p[j];
    }
}

extern "C" void kernel_launch(void* const* d_in, const int* in_sizes, int n_in,
                              void* d_out, int out_size, void* d_ws, size_t ws_size,
                              hipStream_t stream)
{
    const float* x     = (const float*)d_in[0];   // [B, 1024]
    const float* vconf = (const float*)d_in[1];   // [B, 21]
    const float* W     = (const float*)d_in[2];   // [21, 1024]
    const float* bias  = (const float*)d_in[3];   // [21]

    const int Btot = in_sizes[0] / DIM;           // 131072

    float* result = (float*)d_out;                // [B]
    float* out    = (float*)d_out + Btot;         // [B, 21]

    const int blocks = Btot / ROWS_PER_BLOCK;     // 1024
    hipLaunchKernelGGL(linheads_wmma_kernel, dim3(blocks), dim3(256), 0, stream,
                       x, vconf, W, bias, result, out);
}